// umbrella_repsurf_81466939670826
// MI455X (gfx1250) — compile-verified
//
#include <hip/hip_runtime.h>
#include <math.h>

#define B_     4
#define N_     8192
#define KNN_   9
#define CH_    7
#define NPART_ 512

typedef float v2f __attribute__((ext_vector_type(2)));
typedef float v8f __attribute__((ext_vector_type(8)));

// ---- workspace layout (float offsets) ----
#define WS_X4    0                              // B*N float4         = 131072 floats
#define WS_FEAT  131072                         // B*N*9*8            = 2359296 floats
#define WS_PART1 (WS_FEAT + 2359296)            // NPART*14
#define WS_PART2 (WS_PART1 + NPART_ * 14)       // NPART*14
#define WS_ST1   (WS_PART2 + NPART_ * 14)       // 16
#define WS_ST2   (WS_ST1 + 16)                  // 16
// total ~2.51M floats ~ 10 MB

// ------------------------------------------------------------------
// Pass 0: pack x into float4 {x,y,z, -0.5*|x|^2}
// ------------------------------------------------------------------
__global__ __launch_bounds__(256) void prep_kernel(const float* __restrict__ x,
                                                   float4* __restrict__ x4) {
  int i = blockIdx.x * 256 + threadIdx.x;
  if (i < B_ * N_) {
    float a = x[3 * i + 0], b = x[3 * i + 1], c = x[3 * i + 2];
    float4 v;
    v.x = a; v.y = b; v.z = c; v.w = -0.5f * (a * a + b * b + c * c);
    x4[i] = v;
  }
}

// ------------------------------------------------------------------
// Pass 1: fused KNN (WMMA 16x16x4 Gram tiles + in-register top-9)
//         + umbrella feature construction.
// grid = 512 blocks x 64 threads (2 waves); each wave owns 32 query rows.
// ------------------------------------------------------------------
__global__ __launch_bounds__(64) void knn_feat_kernel(const float4* __restrict__ x4,
                                                      float* __restrict__ feat) {
  __shared__ __align__(16) float sc[2][512];   // per-wave 32x16 D-tile scratch

  const int lane = threadIdx.x & 31;
  const int wave = threadIdx.x >> 5;
  const int wg   = blockIdx.x;           // 0..511
  const int b    = wg >> 7;              // batch
  const int r0   = (wg & 127) * 64 + wave * 32;
  const float4* __restrict__ xb = x4 + (size_t)b * N_;

  const int  sub = lane & 15;
  const bool hi  = (lane >= 16);

  // A operands (loop invariant).  A-layout (16x4 f32):
  // lanes 0-15: {K0,K1}; lanes 16-31: {K2,K3}.  K3 == 1.0 carries the bias.
  float4 xa0 = xb[r0 + sub];
  float4 xa1 = xb[r0 + 16 + sub];
  v2f a0, a1;
  a0.x = hi ? xa0.z : xa0.x;  a0.y = hi ? 1.0f : xa0.y;
  a1.x = hi ? xa1.z : xa1.x;  a1.y = hi ? 1.0f : xa1.y;

  float dlist[KNN_]; int ilist[KNN_];
#pragma unroll
  for (int s = 0; s < KNN_; ++s) { dlist[s] = -3.0e38f; ilist[s] = 0; }
  const int myrow = r0 + lane;

  // software-pipelined B tile
  float4 xj = xb[sub];
  for (int j0 = 0; j0 < N_; j0 += 16) {
    const int jn = (j0 + 16 < N_) ? (j0 + 16 + sub) : sub;
    float4 xjn = xb[jn];
    __builtin_prefetch((const void*)&xb[(j0 + 32 < N_) ? (j0 + 32 + sub) : sub], 0, 1);

    // B operand: K3 slot = -0.5*|x_j|^2  ->  D = x_i.x_j - 0.5|x_j|^2
    v2f bv;
    bv.x = hi ? xj.z : xj.x;
    bv.y = hi ? xj.w : xj.y;

    v8f cz = {0.0f, 0.0f, 0.0f, 0.0f, 0.0f, 0.0f, 0.0f, 0.0f};
    v8f d0 = __builtin_amdgcn_wmma_f32_16x16x4_f32(false, a0, false, bv, (short)0, cz, false, false);
    v8f d1 = __builtin_amdgcn_wmma_f32_16x16x4_f32(false, a1, false, bv, (short)0, cz, false, false);

    // spill D tiles to this wave's LDS scratch: row-major 32x16
    const int rb = hi ? 8 : 0;
#pragma unroll
    for (int r = 0; r < 8; ++r) sc[wave][(rb + r) * 16 + sub]      = d0[r];
#pragma unroll
    for (int r = 0; r < 8; ++r) sc[wave][(16 + rb + r) * 16 + sub] = d1[r];
    asm volatile("s_wait_dscnt 0" ::: "memory");

    // each lane scans its own row (16 candidates) from LDS
    const float4* rowp = (const float4*)&sc[wave][lane * 16];
#pragma unroll
    for (int q = 0; q < 4; ++q) {
      float4 mv = rowp[q];
      float m4[4] = {mv.x, mv.y, mv.z, mv.w};
#pragma unroll
      for (int t = 0; t < 4; ++t) {
        const int j = j0 + q * 4 + t;
        float v = m4[t];
        if (j != myrow && v > dlist[KNN_ - 1]) {
          // branch-free ripple insert (v_cndmask chain, no exec juggling)
          int vi = j;
#pragma unroll
          for (int s = 0; s < KNN_; ++s) {
            float od = dlist[s]; int oi = ilist[s];
            bool c = v > od;
            dlist[s] = c ? v : od;
            ilist[s] = c ? vi : oi;
            v        = c ? od : v;
            vi       = c ? oi : vi;
          }
        }
      }
    }
    xj = xjn;
    asm volatile("" ::: "memory");
  }

  // ---- umbrella features for this lane's point ----
  float4 xi = xb[myrow];
  float vx[KNN_], vy[KNN_], vz[KNN_], ph[KNN_];
#pragma unroll
  for (int k = 0; k < KNN_; ++k) {
    float4 xn = xb[ilist[k]];
    vx[k] = xn.x - xi.x;
    vy[k] = xn.y - xi.y;
    vz[k] = xn.z - xi.z;
    ph[k] = atan2f(vy[k], vx[k]);   // monotone in reference's phi
  }
  // unrolled insertion compare-exchange network, ascending phi
#pragma unroll
  for (int i = 1; i < KNN_; ++i) {
#pragma unroll
    for (int j2 = i; j2 > 0; --j2) {
      bool sw = ph[j2 - 1] > ph[j2];
      float p0 = ph[j2 - 1], p1 = ph[j2];
      ph[j2 - 1] = sw ? p1 : p0;  ph[j2] = sw ? p0 : p1;
      float x0 = vx[j2 - 1], x1 = vx[j2];
      vx[j2 - 1] = sw ? x1 : x0;  vx[j2] = sw ? x0 : x1;
      float y0 = vy[j2 - 1], y1 = vy[j2];
      vy[j2 - 1] = sw ? y1 : y0;  vy[j2] = sw ? y0 : y1;
      float z0 = vz[j2 - 1], z1 = vz[j2];
      vz[j2 - 1] = sw ? z1 : z0;  vz[j2] = sw ? z0 : z1;
    }
  }

  // sign from k=0 normal's x component (before normalize; norm > 0)
  float c0x = vy[0] * vz[1] - vz[0] * vy[1] + 1e-5f;
  float sgn = (c0x > 0.0f) ? 1.0f : -1.0f;
  const float is3 = 0.57735026918962576f;  // 1/sqrt(3)

  float* fp = feat + (size_t)(b * N_ + myrow) * (KNN_ * 8);
#pragma unroll
  for (int k = 0; k < KNN_; ++k) {
    int k2 = (k + 1 == KNN_) ? 0 : k + 1;
    float ax = vx[k],  ay = vy[k],  az = vz[k];
    float bx = vx[k2], by = vy[k2], bz = vz[k2];
    float cx = 0.5f * (ax + bx), cy = 0.5f * (ay + by), cz2 = 0.5f * (az + bz);
    float nx = ay * bz - az * by + 1e-5f;
    float ny = az * bx - ax * bz + 1e-5f;
    float nz = ax * by - ay * bx + 1e-5f;
    float inv = sgn / sqrtf(nx * nx + ny * ny + nz * nz);
    nx *= inv; ny *= inv; nz *= inv;
    float pos = (cx * nx + cy * ny + cz2 * nz) * is3;
    float4 f0; f0.x = cx; f0.y = cy; f0.z = cz2; f0.w = nx;
    float4 f1; f1.x = ny; f1.y = nz; f1.z = pos; f1.w = 0.0f;
    ((float4*)fp)[k * 2 + 0] = f0;
    ((float4*)fp)[k * 2 + 1] = f1;
  }
}

// ------------------------------------------------------------------
// BN helpers
// ------------------------------------------------------------------
__device__ __forceinline__ void load_feat7(const float* feat, int r, float f[CH_]) {
  const float4* fp = (const float4*)(feat + (size_t)r * 8);
  float4 f0 = fp[0], f1 = fp[1];
  f[0] = f0.x; f[1] = f0.y; f[2] = f0.z; f[3] = f0.w;
  f[4] = f1.x; f[5] = f1.y; f[6] = f1.z;
}

// ------------------------------------------------------------------
// Pass 2: stats of h1 = feat @ w1^T  (per-channel sum / sumsq partials)
// ------------------------------------------------------------------
__global__ __launch_bounds__(256) void mlp_stats1_kernel(const float* __restrict__ feat,
                                                         const float* __restrict__ w1,
                                                         float* __restrict__ part) {
  __shared__ float red[256][14];
  float s[14];
#pragma unroll
  for (int c = 0; c < 14; ++c) s[c] = 0.0f;

  const int Rtot = B_ * N_ * KNN_;
  for (int r = blockIdx.x * 256 + threadIdx.x; r < Rtot; r += gridDim.x * 256) {
    float f[CH_]; load_feat7(feat, r, f);
#pragma unroll
    for (int o = 0; o < CH_; ++o) {
      float h = 0.0f;
#pragma unroll
      for (int c = 0; c < CH_; ++c) h += f[c] * w1[o * CH_ + c];
      s[o] += h;
      s[CH_ + o] += h * h;
    }
  }
#pragma unroll
  for (int c = 0; c < 14; ++c) red[threadIdx.x][c] = s[c];
  __syncthreads();
  for (int st = 128; st > 0; st >>= 1) {
    if (threadIdx.x < st)
#pragma unroll
      for (int c = 0; c < 14; ++c) red[threadIdx.x][c] += red[threadIdx.x + st][c];
    __syncthreads();
  }
  if (threadIdx.x == 0)
#pragma unroll
    for (int c = 0; c < 14; ++c) part[blockIdx.x * 14 + c] = red[0][c];
}

// ------------------------------------------------------------------
// Reduce partials -> {mean[7], invstd[7]}  (deterministic, 1 block)
// ------------------------------------------------------------------
__global__ void reduce_stats_kernel(const float* __restrict__ part,
                                    float* __restrict__ stats) {
  int c = threadIdx.x;
  if (c < CH_) {
    float s = 0.0f, q = 0.0f;
    for (int g = 0; g < NPART_; ++g) {
      s += part[g * 14 + c];
      q += part[g * 14 + CH_ + c];
    }
    const float cnt = (float)(B_ * N_ * KNN_);
    float m   = s / cnt;
    float var = q / cnt - m * m;
    stats[c]        = m;
    stats[CH_ + c]  = 1.0f / sqrtf(var + 1e-5f);
  }
}

// ------------------------------------------------------------------
// Pass 3: stats of h2 = relu(bn1(h1)) @ w2^T + bias2
// ------------------------------------------------------------------
__global__ __launch_bounds__(256) void mlp_stats2_kernel(const float* __restrict__ feat,
                                                         const float* __restrict__ w1,
                                                         const float* __restrict__ gamma1,
                                                         const float* __restrict__ beta1,
                                                         const float* __restrict__ st1,
                                                         const float* __restrict__ w2,
                                                         const float* __restrict__ bias2,
                                                         float* __restrict__ part) {
  __shared__ float red[256][14];
  float s[14];
#pragma unroll
  for (int c = 0; c < 14; ++c) s[c] = 0.0f;

  const int Rtot = B_ * N_ * KNN_;
  for (int r = blockIdx.x * 256 + threadIdx.x; r < Rtot; r += gridDim.x * 256) {
    float f[CH_]; load_feat7(feat, r, f);
    float a1[CH_];
#pragma unroll
    for (int o = 0; o < CH_; ++o) {
      float h = 0.0f;
#pragma unroll
      for (int c = 0; c < CH_; ++c) h += f[c] * w1[o * CH_ + c];
      h = (h - st1[o]) * st1[CH_ + o] * gamma1[o] + beta1[o];
      a1[o] = h > 0.0f ? h : 0.0f;
    }
#pragma unroll
    for (int o = 0; o < CH_; ++o) {
      float h = bias2[o];
#pragma unroll
      for (int c = 0; c < CH_; ++c) h += a1[c] * w2[o * CH_ + c];
      s[o] += h;
      s[CH_ + o] += h * h;
    }
  }
#pragma unroll
  for (int c = 0; c < 14; ++c) red[threadIdx.x][c] = s[c];
  __syncthreads();
  for (int st = 128; st > 0; st >>= 1) {
    if (threadIdx.x < st)
#pragma unroll
      for (int c = 0; c < 14; ++c) red[threadIdx.x][c] += red[threadIdx.x + st][c];
    __syncthreads();
  }
  if (threadIdx.x == 0)
#pragma unroll
    for (int c = 0; c < 14; ++c) part[blockIdx.x * 14 + c] = red[0][c];
}

// ------------------------------------------------------------------
// Pass 4: full MLP + max-pool over k + concat x  -> out (B,N,10)
// ------------------------------------------------------------------
__global__ __launch_bounds__(256) void final_kernel(const float4* __restrict__ x4,
                                                    const float* __restrict__ feat,
                                                    const float* __restrict__ w1,
                                                    const float* __restrict__ gamma1,
                                                    const float* __restrict__ beta1,
                                                    const float* __restrict__ st1,
                                                    const float* __restrict__ w2,
                                                    const float* __restrict__ bias2,
                                                    const float* __restrict__ gamma2,
                                                    const float* __restrict__ beta2,
                                                    const float* __restrict__ st2,
                                                    const float* __restrict__ w3,
                                                    const float* __restrict__ bias3,
                                                    float* __restrict__ out) {
  int p = blockIdx.x * 256 + threadIdx.x;
  if (p >= B_ * N_) return;

  float pooled[CH_];
#pragma unroll
  for (int o = 0; o < CH_; ++o) pooled[o] = -3.0e38f;

  for (int k = 0; k < KNN_; ++k) {
    int r = p * KNN_ + k;
    float f[CH_]; load_feat7(feat, r, f);
    float a1[CH_], a2[CH_];
#pragma unroll
    for (int o = 0; o < CH_; ++o) {
      float h = 0.0f;
#pragma unroll
      for (int c = 0; c < CH_; ++c) h += f[c] * w1[o * CH_ + c];
      h = (h - st1[o]) * st1[CH_ + o] * gamma1[o] + beta1[o];
      a1[o] = h > 0.0f ? h : 0.0f;
    }
#pragma unroll
    for (int o = 0; o < CH_; ++o) {
      float h = bias2[o];
#pragma unroll
      for (int c = 0; c < CH_; ++c) h += a1[c] * w2[o * CH_ + c];
      h = (h - st2[o]) * st2[CH_ + o] * gamma2[o] + beta2[o];
      a2[o] = h > 0.0f ? h : 0.0f;
    }
#pragma unroll
    for (int o = 0; o < CH_; ++o) {
      float h = bias3[o];
#pragma unroll
      for (int c = 0; c < CH_; ++c) h += a2[c] * w3[o * CH_ + c];
      pooled[o] = h > pooled[o] ? h : pooled[o];
    }
  }

  float4 xi = x4[p];
  float* op = out + (size_t)p * 10;
  op[0] = xi.x; op[1] = xi.y; op[2] = xi.z;
#pragma unroll
  for (int o = 0; o < CH_; ++o) op[3 + o] = pooled[o];
}

// ------------------------------------------------------------------
extern "C" void kernel_launch(void* const* d_in, const int* in_sizes, int n_in,
                              void* d_out, int out_size, void* d_ws, size_t ws_size,
                              hipStream_t stream) {
  (void)in_sizes; (void)n_in; (void)out_size; (void)ws_size;

  const float* x      = (const float*)d_in[0];
  const float* w1     = (const float*)d_in[1];
  const float* gamma1 = (const float*)d_in[2];
  const float* beta1  = (const float*)d_in[3];
  const float* w2     = (const float*)d_in[4];
  const float* bias2  = (const float*)d_in[5];
  const float* gamma2 = (const float*)d_in[6];
  const float* beta2  = (const float*)d_in[7];
  const float* w3     = (const float*)d_in[8];
  const float* bias3  = (const float*)d_in[9];

  float*  ws    = (float*)d_ws;
  float4* x4    = (float4*)(ws + WS_X4);
  float*  feat  = ws + WS_FEAT;
  float*  part1 = ws + WS_PART1;
  float*  part2 = ws + WS_PART2;
  float*  st1   = ws + WS_ST1;
  float*  st2   = ws + WS_ST2;
  float*  out   = (float*)d_out;

  prep_kernel<<<(B_ * N_ + 255) / 256, 256, 0, stream>>>(x, x4);
  knn_feat_kernel<<<512, 64, 0, stream>>>(x4, feat);
  mlp_stats1_kernel<<<NPART_, 256, 0, stream>>>(feat, w1, part1);
  reduce_stats_kernel<<<1, 32, 0, stream>>>(part1, st1);
  mlp_stats2_kernel<<<NPART_, 256, 0, stream>>>(feat, w1, gamma1, beta1, st1, w2, bias2, part2);
  reduce_stats_kernel<<<1, 32, 0, stream>>>(part2, st2);
  final_kernel<<<128, 256, 0, stream>>>(x4, feat, w1, gamma1, beta1, st1,
                                        w2, bias2, gamma2, beta2, st2, w3, bias3, out);
}